// GraphAPPNP_63015760166992
// MI455X (gfx1250) — compile-verified
//
#include <hip/hip_runtime.h>
#include <cstddef>
#include <cstdint>

// GCN + APPNP on MI455X (gfx1250).
// Propagates are L2-bandwidth bound (whole working set < 192MB L2); GEMMs are
// tiny (~4 GFLOP) so we keep them full fp32 via V_WMMA_F32_16X16X4_F32.

#define APPNP_ALPHA 0.1f
#define APPNP_K 10

typedef __attribute__((ext_vector_type(2))) float v2f;
typedef __attribute__((ext_vector_type(8))) float v8f;

#if defined(__gfx1250__) && __has_builtin(__builtin_amdgcn_wmma_f32_16x16x4_f32)
#define HAVE_WMMA_F32 1
#else
#define HAVE_WMMA_F32 0
#endif

__device__ __forceinline__ void atomAddF(float* p, float v) {
    // agent-scope relaxed fp32 add -> global_atomic_add_f32 (no CAS loop)
    __hip_atomic_fetch_add(p, v, __ATOMIC_RELAXED, __HIP_MEMORY_SCOPE_AGENT);
}

// ---------------- degree / normalization ----------------

__global__ void count_deg_kernel(const int* __restrict__ col, float* __restrict__ deg, int E) {
    int i = blockIdx.x * blockDim.x + threadIdx.x;
    if (i < E) atomAddF(&deg[col[i]], 1.0f);  // counts exact in fp32 (< 2^24)
}

__global__ void make_dinv_kernel(float* __restrict__ d, int N) {
    int i = blockIdx.x * blockDim.x + threadIdx.x;
    if (i < N) d[i] = rsqrtf(d[i] + 1.0f);  // +1 self loop; always > 0
}

// ---------------- WMMA fp32 GEMM: C[M,N] = A[M,K] @ B[K,N] ----------------
// One 16x16 output tile per wave32. K stepped by 4 (V_WMMA_F32_16X16X4_F32).
// A 16x4 lane layout (ISA 7.12.2): lanes0-15 M=lane,K={0,1}; lanes16-31 M=lane-16,K={2,3}.
// B 4x16: rows striped across lanes within a VGPR (mirror of C layout).
// C 16x16: VGPR j -> row j + (lane>>4)*8, col = lane&15.
__global__ void gemm_wmma_f32(const float* __restrict__ A, const float* __restrict__ B,
                              float* __restrict__ C, int M, int N, int K) {
    const int lane = threadIdx.x & 31;
    const int wid  = (blockIdx.x * blockDim.x + threadIdx.x) >> 5;
    const int ntN  = N >> 4;
    const int tM   = wid / ntN;
    const int tN   = wid - tM * ntN;
    if (tM * 16 >= M) return;  // wave-uniform: EXEC stays all-ones for WMMA

    const int mrow = tM * 16 + (lane & 15);
    const int ncol = tN * 16 + (lane & 15);
    const int kf   = (lane >> 4) << 1;       // 0 or 2
    const int rb   = tM * 16 + ((lane >> 4) << 3);

#if HAVE_WMMA_F32
    v8f acc = {0.f, 0.f, 0.f, 0.f, 0.f, 0.f, 0.f, 0.f};
    const float* ap = A + (size_t)mrow * K;
    for (int k = 0; k < K; k += 4) {
        v2f a, b;
        a.x = ap[k + kf];
        a.y = ap[k + kf + 1];
        b.x = B[(size_t)(k + kf) * N + ncol];
        b.y = B[(size_t)(k + kf + 1) * N + ncol];
        acc = __builtin_amdgcn_wmma_f32_16x16x4_f32(
            /*neg_a=*/false, a, /*neg_b=*/false, b,
            /*c_mod=*/(short)0, acc, /*reuse_a=*/false, /*reuse_b=*/false);
    }
#pragma unroll
    for (int j = 0; j < 8; ++j)
        C[(size_t)(rb + j) * N + ncol] = acc[j];
#else
    // scalar fallback (host pass / missing builtin); same output mapping
    float acc[8] = {0.f, 0.f, 0.f, 0.f, 0.f, 0.f, 0.f, 0.f};
    for (int k = 0; k < K; ++k) {
        float bv = B[(size_t)k * N + ncol];
        for (int j = 0; j < 8; ++j)
            acc[j] += A[(size_t)(rb + j) * K + k] * bv;
    }
    for (int j = 0; j < 8; ++j)
        C[(size_t)(rb + j) * N + ncol] = acc[j];
    (void)mrow; (void)kf;
#endif
}

// ---------------- edge scatter: acc[col] += dinv[row]*dinv[col]*h[row] ----------------
// One wave per edge; lane l handles float4 chunk l of the F-wide feature row.
template <int F>
__global__ void edge_scatter_kernel(const int* __restrict__ row, const int* __restrict__ col,
                                    const float* __restrict__ dinv, const float* __restrict__ h,
                                    float* __restrict__ acc, int E) {
    const int lane = threadIdx.x & 31;
    const int wave = (blockIdx.x * blockDim.x + threadIdx.x) >> 5;
    const int nw   = (gridDim.x * blockDim.x) >> 5;
    constexpr int C4 = F / 4;  // 32 (F=128) or 16 (F=64)
    for (int e = wave; e < E; e += nw) {
        const int r = row[e];
        const int c = col[e];
        if (e + nw < E)  // warm L2/L0 for the next gather row -> global_prefetch_b8
            __builtin_prefetch(h + (size_t)row[e + nw] * F, 0, 1);
        const float w = dinv[r] * dinv[c];
        if (lane < C4) {
            const float4 v = reinterpret_cast<const float4*>(h + (size_t)r * F)[lane];
            float* o = acc + (size_t)c * F + lane * 4;
            atomAddF(o + 0, w * v.x);
            atomAddF(o + 1, w * v.y);
            atomAddF(o + 2, w * v.z);
            atomAddF(o + 3, w * v.w);
        }
    }
}

// ---------------- per-node combine kernels (self-loop term folded in) ----------------

// layer1: h0 = h = relu(tmp + dinv^2*hpre + b1)
__global__ void combine_layer1_kernel(const float* __restrict__ tmp, const float* __restrict__ hpre,
                                      const float* __restrict__ dinv, const float* __restrict__ b1,
                                      float* __restrict__ h0, float* __restrict__ h,
                                      int total, int F) {
    int idx = blockIdx.x * blockDim.x + threadIdx.x;
    if (idx >= total) return;
    const int i = idx / F;
    const int f = idx - i * F;
    const float di = dinv[i];
    float v = tmp[idx] + di * di * hpre[idx] + b1[f];
    v = fmaxf(v, 0.0f);
    h0[idx] = v;
    h[idx]  = v;
}

// APPNP: nxt = (1-alpha)*(tmp + dinv^2*cur) + alpha*h0
__global__ void combine_appnp_kernel(const float* __restrict__ tmp, const float* __restrict__ cur,
                                     const float* __restrict__ h0, const float* __restrict__ dinv,
                                     float* __restrict__ nxt, int total, int F) {
    int idx = blockIdx.x * blockDim.x + threadIdx.x;
    if (idx >= total) return;
    const int i = idx / F;
    const float di = dinv[i];
    nxt[idx] = (1.0f - APPNP_ALPHA) * (tmp[idx] + di * di * cur[idx]) + APPNP_ALPHA * h0[idx];
}

// final: out += dinv^2*g + b2   (out already holds the edge-scatter sum)
__global__ void combine_final_kernel(float* __restrict__ out, const float* __restrict__ g,
                                     const float* __restrict__ dinv, const float* __restrict__ b2,
                                     int total, int F) {
    int idx = blockIdx.x * blockDim.x + threadIdx.x;
    if (idx >= total) return;
    const int i = idx / F;
    const int f = idx - i * F;
    const float di = dinv[i];
    out[idx] = out[idx] + di * di * g[idx] + b2[f];
}

// ---------------- launch ----------------

extern "C" void kernel_launch(void* const* d_in, const int* in_sizes, int n_in,
                              void* d_out, int out_size, void* d_ws, size_t ws_size,
                              hipStream_t stream) {
    (void)n_in; (void)out_size; (void)ws_size;
    const float* x  = (const float*)d_in[0];   // [N,256]
    const int*   ei = (const int*)d_in[1];     // [2,E]
    const float* W1 = (const float*)d_in[2];   // [256,128]
    const float* b1 = (const float*)d_in[3];   // [128]
    const float* W2 = (const float*)d_in[4];   // [128,64]
    const float* b2 = (const float*)d_in[5];   // [64]
    float* out = (float*)d_out;                // [N,64]

    const int N = in_sizes[0] / 256;           // 50000
    const int E = in_sizes[1] / 2;             // 800000
    const int* row = ei;
    const int* col = ei + E;

    // workspace layout (floats): dinv | bufA(=hpre, later ping) | h0 | bufB | tmp
    float* p    = (float*)d_ws;
    float* dinv = p;  p += N;
    float* bufA = p;  p += (size_t)N * 128;    // hpre, then ping-pong
    float* h0   = p;  p += (size_t)N * 128;
    float* bufB = p;  p += (size_t)N * 128;
    float* tmp  = p;  p += (size_t)N * 128;
    float* g    = tmp;                          // reuse: tmp dead once APPNP loop ends

    const int total128 = N * 128;
    const int total64  = N * 64;

    // 1) degree + dinv
    hipMemsetAsync(dinv, 0, (size_t)N * sizeof(float), stream);
    count_deg_kernel<<<(E + 255) / 256, 256, 0, stream>>>(col, dinv, E);
    make_dinv_kernel<<<(N + 255) / 256, 256, 0, stream>>>(dinv, N);

    // 2) hpre = x @ W1   (N x 256 x 128), one 16x16 tile per wave, 4 waves/block
    {
        int waves = (N / 16) * (128 / 16);
        gemm_wmma_f32<<<(waves + 3) / 4, 128, 0, stream>>>(x, W1, bufA, N, 128, 256);
    }

    // 3) GCN propagate + bias + relu -> h0, cur
    hipMemsetAsync(tmp, 0, (size_t)total128 * sizeof(float), stream);
    edge_scatter_kernel<128><<<2048, 256, 0, stream>>>(row, col, dinv, bufA, tmp, E);
    combine_layer1_kernel<<<(total128 + 255) / 256, 256, 0, stream>>>(
        tmp, bufA, dinv, b1, h0, bufB, total128, 128);

    // 4) APPNP iterations (ping-pong bufB <-> bufA; bufA's hpre contents are dead now)
    float* cur = bufB;
    float* nxt = bufA;
    for (int k = 0; k < APPNP_K; ++k) {
        hipMemsetAsync(tmp, 0, (size_t)total128 * sizeof(float), stream);
        edge_scatter_kernel<128><<<2048, 256, 0, stream>>>(row, col, dinv, cur, tmp, E);
        combine_appnp_kernel<<<(total128 + 255) / 256, 256, 0, stream>>>(
            tmp, cur, h0, dinv, nxt, total128, 128);
        float* t = cur; cur = nxt; nxt = t;
    }

    // 5) g = cur @ W2   (N x 128 x 64); g aliases tmp (free now)
    {
        int waves = (N / 16) * (64 / 16);
        gemm_wmma_f32<<<(waves + 3) / 4, 128, 0, stream>>>(cur, W2, g, N, 64, 128);
    }

    // 6) final propagate into d_out, then add self-loop term + bias
    hipMemsetAsync(out, 0, (size_t)total64 * sizeof(float), stream);
    edge_scatter_kernel<64><<<2048, 256, 0, stream>>>(row, col, dinv, g, out, E);
    combine_final_kernel<<<(total64 + 255) / 256, 256, 0, stream>>>(
        out, g, dinv, b2, total64, 64);
}